// RoIFeatureExtractor_43920335569143
// MI455X (gfx1250) — compile-verified
//
#include <hip/hip_runtime.h>
#include <hip/hip_bf16.h>

#define B 2
#define N 4096
#define C 256
#define K 64
#define MROWS (B * K)   // 128 rows into the MLP

typedef __attribute__((ext_vector_type(2))) float v2f;
typedef __attribute__((ext_vector_type(8))) float v8f;

// ---------------------------------------------------------------------------
// Kernel 1: per-(b,k) box-masked max pool over point features.
// One 256-thread workgroup per (b,k). Phase 1 builds an LDS list of in-box
// point indices (compacted via LDS atomic append). Phase 2: thread c owns
// channel c and max-reduces over the list with fully coalesced 1KB row reads.
// ---------------------------------------------------------------------------
__global__ void __launch_bounds__(256)
roi_maxpool_kernel(const float* __restrict__ points,
                   const float* __restrict__ feats,
                   const float* __restrict__ proposals,
                   float* __restrict__ pooled)
{
    __shared__ int s_idx[N];
    __shared__ int s_cnt;

    const int bk = blockIdx.x;          // 0..127
    const int b  = bk / K;
    const int tid = threadIdx.x;

    if (tid == 0) s_cnt = 0;
    __syncthreads();

    // Box bounds (reference ignores ry for the containment test)
    const float* prop = proposals + (size_t)bk * 7;
    const float cx = prop[0], cy = prop[1], cz = prop[2];
    const float hx = prop[3] * 0.5f, hy = prop[4] * 0.5f, hz = prop[5] * 0.5f;
    const float lox = cx - hx, hix = cx + hx;
    const float loy = cy - hy, hiy = cy + hy;
    const float loz = cz - hz, hiz = cz + hz;

    const float* pts = points + (size_t)b * N * 3;

    // Phase 1: compact in-box indices into LDS
    for (int n = tid; n < N; n += 256) {
        const float px = pts[n * 3 + 0];
        const float py = pts[n * 3 + 1];
        const float pz = pts[n * 3 + 2];
        const bool inside = (px > lox) && (px < hix) &&
                            (py > loy) && (py < hiy) &&
                            (pz > loz) && (pz < hiz);
        if (inside) {
            int pos = atomicAdd(&s_cnt, 1);
            s_idx[pos] = n;
        }
    }
    __syncthreads();

    const int cnt = s_cnt;

    // Phase 2: channel-parallel max over the gathered list
    const float* fb = feats + (size_t)b * N * C;
    const int c = tid;                  // C == blockDim.x == 256
    float best = -__builtin_inff();
    for (int i = 0; i < cnt; ++i) {
        const int n = s_idx[i];
        best = fmaxf(best, fb[(size_t)n * C + c]);
    }
    pooled[(size_t)bk * C + c] = (cnt > 0) ? best : 0.0f;
}

// ---------------------------------------------------------------------------
// Kernel 2/3: fused GEMM + bias + ReLU using V_WMMA_F32_16X16X4_F32.
//   out[M x 256] = relu(A[M x 256] @ W[256 x 256] + bias)
// Grid: 16 blocks x 256 threads = 128 wave32 waves; each wave computes one
// 16x16 output tile, accumulating 64 fp32 WMMAs over K=256 in steps of 4.
//
// Fragment layouts (ISA 7.12.2):
//   A 16x4 : lane L, VGPR v -> A[mt*16 + L%16][k0 + 2*(L/16) + v]  (float2)
//   B 4x16 : lane L, VGPR v -> W[k0 + 2*(L/16) + v][nt*16 + L%16]
//   C/D    : lane L, VGPR v -> row mt*16 + v + 8*(L/16), col nt*16 + L%16
// ---------------------------------------------------------------------------
__global__ void __launch_bounds__(256)
gemm_bias_relu_wmma_kernel(const float* __restrict__ A,
                           const float* __restrict__ W,
                           const float* __restrict__ bias,
                           float* __restrict__ out)
{
    const int lane  = threadIdx.x & 31;
    const int wave  = threadIdx.x >> 5;             // 0..7
    const int tile  = blockIdx.x * 8 + wave;        // 0..127
    const int mt    = tile >> 4;                    // 0..7  (M = 128)
    const int nt    = tile & 15;                    // 0..15 (Ncols = 256)

    const int half = lane >> 4;                     // 0 or 1
    const int l16  = lane & 15;
    const int m    = mt * 16 + l16;
    const int n    = nt * 16 + l16;

    const float* arow = A + (size_t)m * C;

    v8f acc = {};
    for (int k0 = 0; k0 < C; k0 += 4) {
        const int ka = k0 + half * 2;
        // A fragment: two consecutive k elements of row m (contiguous load)
        v2f afrag;
        afrag.x = arow[ka + 0];
        afrag.y = arow[ka + 1];
        // B fragment: rows ka, ka+1 of W at column n
        v2f bfrag;
        bfrag.x = W[(size_t)(ka + 0) * 256 + n];
        bfrag.y = W[(size_t)(ka + 1) * 256 + n];
        acc = __builtin_amdgcn_wmma_f32_16x16x4_f32(
            /*neg_a=*/false, afrag,
            /*neg_b=*/false, bfrag,
            /*c_mod=*/(short)0, acc,
            /*reuse_a=*/false, /*reuse_b=*/false);
    }

    const float bn = bias[n];
#pragma unroll
    for (int v = 0; v < 8; ++v) {
        const int row = mt * 16 + v + half * 8;
        const float val = fmaxf(acc[v] + bn, 0.0f);
        out[(size_t)row * 256 + n] = val;
    }
}

// ---------------------------------------------------------------------------
// Launch: pool -> ws.pooled ; gemm1 -> ws.h ; gemm2 -> d_out
// Inputs: 0=points 1=point_features 2=proposals 3=W1 4=b1 5=W2 6=b2
// ---------------------------------------------------------------------------
extern "C" void kernel_launch(void* const* d_in, const int* in_sizes, int n_in,
                              void* d_out, int out_size, void* d_ws, size_t ws_size,
                              hipStream_t stream)
{
    (void)in_sizes; (void)n_in; (void)out_size; (void)ws_size;

    const float* points    = (const float*)d_in[0];
    const float* feats     = (const float*)d_in[1];
    const float* proposals = (const float*)d_in[2];
    const float* W1        = (const float*)d_in[3];
    const float* b1        = (const float*)d_in[4];
    const float* W2        = (const float*)d_in[5];
    const float* b2        = (const float*)d_in[6];

    float* pooled = (float*)d_ws;                    // 128 x 256 fp32
    float* h      = pooled + (size_t)MROWS * 256;    // 128 x 256 fp32
    float* outp   = (float*)d_out;                   // 128 x 256 fp32

    roi_maxpool_kernel<<<B * K, 256, 0, stream>>>(points, feats, proposals, pooled);
    gemm_bias_relu_wmma_kernel<<<16, 256, 0, stream>>>(pooled, W1, b1, h);
    gemm_bias_relu_wmma_kernel<<<16, 256, 0, stream>>>(h, W2, b2, outp);
}